// Model_Learning_35150012351299
// MI455X (gfx1250) — compile-verified
//
#include <hip/hip_runtime.h>

// CDNA5 / gfx1250 persistent Heun-RK2 MLP integrator using V_WMMA_F32_16X16X4_F32.
// One workgroup owns a 16-row batch tile and integrates all 256 steps in LDS.
// Weights are repacked once per launch into per-lane WMMA fragment order so every
// B fragment is a single contiguous global_load_b64.

typedef __attribute__((ext_vector_type(2))) float v2f;
typedef __attribute__((ext_vector_type(8))) float v8f;

#define DT_F    0.1f
#define BATCH   512
#define NFEAT   256
#define TSTEPS  256
#define HID     1024
#define ROWS    16
#define THREADS 512          // 16 waves of 32

// LDS row strides, padded so stride % 8 == 4 -> half-waves (8 rows apart) hit
// different banks (8*stride % 64 == 32).
#define SX 260               // 256-wide buffers (X, XT, K1)
#define SH 1028              // 1024-wide buffers (H1, H2)

// ---------------------------------------------------------------------------
// Weight repack: Wp[tile][kstep][lane][2], so that lane L of a wave loads its
// B fragment (rows k0+2*(L>=16)+{0,1}, col n0+(L&15)) as ONE b64 from
//   Wp + (tile*(K/4) + kstep)*64 + L*2.
// ---------------------------------------------------------------------------
__global__ void repack_weights(const float* __restrict__ W, float* __restrict__ Wp,
                               int K, int Nout)
{
  const int total = (Nout / 16) * (K / 4) * 32;
  const int idx = blockIdx.x * blockDim.x + threadIdx.x;
  if (idx >= total) return;
  const int lane = idx & 31;
  const int kt   = (idx >> 5) % (K / 4);
  const int tile = (idx >> 5) / (K / 4);
  const int half = lane >> 4;
  const int l16  = lane & 15;
  const int krow = 4 * kt + 2 * half;
  const int col  = tile * 16 + l16;
  Wp[(size_t)idx * 2 + 0] = W[(size_t)krow * Nout + col];
  Wp[(size_t)idx * 2 + 1] = W[(size_t)(krow + 1) * Nout + col];
}

// ---------------------------------------------------------------------------
// Four 16x16 output tiles per wave (n0 .. n0+48), one shared A fragment,
// 4 independent WMMA accumulator chains to hide XDL latency.
// A fragment (16x4 f32): lanes 0-15 -> K=k0,k0+1 ; lanes 16-31 -> K=k0+2,k0+3.
// C/D tile: VGPR i, lane L -> row = i + 8*(L>=16), col = L&15.
// ---------------------------------------------------------------------------
template<int K, bool RELU>
__device__ __forceinline__ void gemm_quad(
    const float* __restrict__ A, int lda,
    const float* __restrict__ Wp,       // repacked
    const float* __restrict__ bias,
    float* __restrict__ Out, int ldo,
    int n0, int lane)
{
  const int half = lane >> 4;
  const int l16  = lane & 15;
  v8f acc0 = {}, acc1 = {}, acc2 = {}, acc3 = {};
  const float* ap = A + l16 * lda + 2 * half;
  const int t0 = n0 >> 4;
  const float* bp0 = Wp + ((size_t)(t0 + 0) * (K / 4)) * 64 + lane * 2;
  const float* bp1 = Wp + ((size_t)(t0 + 1) * (K / 4)) * 64 + lane * 2;
  const float* bp2 = Wp + ((size_t)(t0 + 2) * (K / 4)) * 64 + lane * 2;
  const float* bp3 = Wp + ((size_t)(t0 + 3) * (K / 4)) * 64 + lane * 2;
  __builtin_prefetch(bp0, 0, 0);       // global_prefetch_b8; streams are L2-hot
  __builtin_prefetch(bp1, 0, 0);
  __builtin_prefetch(bp2, 0, 0);
  __builtin_prefetch(bp3, 0, 0);
  #pragma unroll 4
  for (int k = 0; k < K; k += 4) {
    v2f a;
    a.x = ap[0]; a.y = ap[1];                       // LDS (ds_load_2addr)
    const v2f b0 = *(const v2f*)bp0;                // single global_load_b64 each
    const v2f b1 = *(const v2f*)bp1;
    const v2f b2 = *(const v2f*)bp2;
    const v2f b3 = *(const v2f*)bp3;
    acc0 = __builtin_amdgcn_wmma_f32_16x16x4_f32(false, a, false, b0, (short)0, acc0, false, false);
    acc1 = __builtin_amdgcn_wmma_f32_16x16x4_f32(false, a, false, b1, (short)0, acc1, false, false);
    acc2 = __builtin_amdgcn_wmma_f32_16x16x4_f32(false, a, false, b2, (short)0, acc2, false, false);
    acc3 = __builtin_amdgcn_wmma_f32_16x16x4_f32(false, a, false, b3, (short)0, acc3, false, false);
    ap  += 4;
    bp0 += 64; bp1 += 64; bp2 += 64; bp3 += 64;
  }
  const float bs0 = bias[n0      + l16];
  const float bs1 = bias[n0 + 16 + l16];
  const float bs2 = bias[n0 + 32 + l16];
  const float bs3 = bias[n0 + 48 + l16];
  #pragma unroll
  for (int i = 0; i < 8; ++i) {
    const int row = i + 8 * half;
    float v0 = acc0[i] + bs0;
    float v1 = acc1[i] + bs1;
    float v2 = acc2[i] + bs2;
    float v3 = acc3[i] + bs3;
    if (RELU) {
      v0 = fmaxf(v0, 0.0f); v1 = fmaxf(v1, 0.0f);
      v2 = fmaxf(v2, 0.0f); v3 = fmaxf(v3, 0.0f);
    }
    Out[row * ldo + n0      + l16] = v0;
    Out[row * ldo + n0 + 16 + l16] = v1;
    Out[row * ldo + n0 + 32 + l16] = v2;
    Out[row * ldo + n0 + 48 + l16] = v3;
  }
}

// One tile per wave, K split in halves -> 2 independent accumulator chains.
template<int K, bool RELU>
__device__ __forceinline__ void gemm_single_ksplit(
    const float* __restrict__ A, int lda,
    const float* __restrict__ Wp,
    const float* __restrict__ bias,
    float* __restrict__ Out, int ldo,
    int n0, int lane)
{
  const int half = lane >> 4;
  const int l16  = lane & 15;
  v8f accA = {}, accB = {};
  const float* apA = A + l16 * lda + 2 * half;
  const float* apB = apA + (K / 2);
  const int t0 = n0 >> 4;
  const float* bpA = Wp + ((size_t)t0 * (K / 4)) * 64 + lane * 2;
  const float* bpB = bpA + (size_t)(K / 8) * 64;
  __builtin_prefetch(bpA, 0, 0);
  __builtin_prefetch(bpB, 0, 0);
  #pragma unroll 4
  for (int k = 0; k < K / 2; k += 4) {
    v2f aA, aB;
    aA.x = apA[0]; aA.y = apA[1];
    aB.x = apB[0]; aB.y = apB[1];
    const v2f bA = *(const v2f*)bpA;
    const v2f bB = *(const v2f*)bpB;
    accA = __builtin_amdgcn_wmma_f32_16x16x4_f32(false, aA, false, bA, (short)0, accA, false, false);
    accB = __builtin_amdgcn_wmma_f32_16x16x4_f32(false, aB, false, bB, (short)0, accB, false, false);
    apA += 4; apB += 4;
    bpA += 64; bpB += 64;
  }
  const float bs = bias[n0 + l16];
  #pragma unroll
  for (int i = 0; i < 8; ++i) {
    const int row = i + 8 * half;
    float v = accA[i] + accB[i] + bs;
    if (RELU) v = fmaxf(v, 0.0f);
    Out[row * ldo + n0 + l16] = v;
  }
}

// Full 3-layer MLP on a 16-row tile. 16 waves: hidden layers -> 4 N-tiles per
// wave (one quad call), output layer -> 1 N-tile per wave (K split in 2).
__device__ __forceinline__ void mlp_16rows(
    const float* __restrict__ Xin,
    float* __restrict__ H1, float* __restrict__ H2, float* __restrict__ Out,
    const float* __restrict__ Wp1, const float* __restrict__ b1,
    const float* __restrict__ Wp2, const float* __restrict__ b2,
    const float* __restrict__ Wp3, const float* __restrict__ b3,
    int wave, int lane)
{
  const int n0 = wave * 64;  // 16 waves * 64 cols = 1024
  gemm_quad<NFEAT, true>(Xin, SX, Wp1, b1, H1, SH, n0, lane);
  __syncthreads();
  gemm_quad<HID, true>(H1, SH, Wp2, b2, H2, SH, n0, lane);
  __syncthreads();
  gemm_single_ksplit<HID, false>(H2, SH, Wp3, b3, Out, SX, wave * 16, lane);
  __syncthreads();
}

__global__ __launch_bounds__(THREADS, 1)
void heun_mlp_persistent(const float* __restrict__ Input,
                         const float* __restrict__ y_true,
                         const float* __restrict__ Wp1, const float* __restrict__ b1,
                         const float* __restrict__ Wp2, const float* __restrict__ b2,
                         const float* __restrict__ Wp3, const float* __restrict__ b3,
                         float* __restrict__ err_partials,
                         float* __restrict__ y_out)
{
  extern __shared__ float smem[];
  float* X  = smem;                 // 16 x SX  : current state
  float* XT = X  + ROWS * SX;       // 16 x SX  : X + DT*k1, then k2
  float* K1 = XT + ROWS * SX;       // 16 x SX  : k1
  float* H1 = K1 + ROWS * SX;       // 16 x SH  : hidden 1
  float* H2 = H1 + ROWS * SH;       // 16 x SH  : hidden 2

  const int tid  = threadIdx.x;
  const int lane = tid & 31;
  const int wave = tid >> 5;
  const int b0   = blockIdx.x * ROWS;

  // X = Input[:, :, 0] for this row tile
  for (int i = tid; i < ROWS * NFEAT; i += THREADS) {
    const int r = i / NFEAT, n = i % NFEAT;
    X[r * SX + n] = Input[((size_t)(b0 + r) * NFEAT + n) * TSTEPS];
  }
  __syncthreads();

  float sse = 0.0f;

  for (int t = 0; t < TSTEPS; ++t) {
    // k1 = f(X)
    mlp_16rows(X, H1, H2, K1, Wp1, b1, Wp2, b2, Wp3, b3, wave, lane);

    // XT = X + DT*k1
    for (int i = tid; i < ROWS * NFEAT; i += THREADS) {
      const int r = i / NFEAT, n = i % NFEAT;
      XT[r * SX + n] = X[r * SX + n] + DT_F * K1[r * SX + n];
    }
    __syncthreads();

    // k2 = f(XT)  (output overwrites XT: layer-1 reads of XT complete before
    // layer-3 writes thanks to the intra-MLP barriers)
    mlp_16rows(XT, H1, H2, XT, Wp1, b1, Wp2, b2, Wp3, b3, wave, lane);

    // X += 0.5*DT*(k1 + k2); emit y[:, :, t]; accumulate squared error
    for (int i = tid; i < ROWS * NFEAT; i += THREADS) {
      const int r = i / NFEAT, n = i % NFEAT;
      const float xn = X[r * SX + n] + 0.5f * DT_F * (K1[r * SX + n] + XT[r * SX + n]);
      X[r * SX + n] = xn;
      const size_t oidx = ((size_t)(b0 + r) * NFEAT + n) * TSTEPS + t;
      y_out[oidx] = xn;
      const float d = y_true[oidx] - xn;
      sse += d * d;
    }
    __syncthreads();
  }

  // Block-reduce SSE (reuse H1 region) -> one partial per block
  H1[tid] = sse;
  __syncthreads();
  for (int s = THREADS / 2; s > 0; s >>= 1) {
    if (tid < s) H1[tid] += H1[tid + s];
    __syncthreads();
  }
  if (tid == 0) err_partials[blockIdx.x] = H1[0];
}

__global__ void finalize_err(const float* __restrict__ partials, int n,
                             float* __restrict__ err_out, float inv_count)
{
  if (threadIdx.x == 0) {
    float s = 0.0f;
    for (int i = 0; i < n; ++i) s += partials[i];
    err_out[0] = s * inv_count;
  }
}

extern "C" void kernel_launch(void* const* d_in, const int* in_sizes, int n_in,
                              void* d_out, int out_size, void* d_ws, size_t ws_size,
                              hipStream_t stream)
{
  (void)in_sizes; (void)n_in; (void)out_size; (void)ws_size;
  const float* Input  = (const float*)d_in[0];
  const float* y_true = (const float*)d_in[1];
  // d_in[2] = t0s (unused by the math)
  const float* W1 = (const float*)d_in[3];
  const float* b1 = (const float*)d_in[4];
  const float* W2 = (const float*)d_in[5];
  const float* b2 = (const float*)d_in[6];
  const float* W3 = (const float*)d_in[7];
  const float* b3 = (const float*)d_in[8];

  float* out = (float*)d_out;               // [0] = err, [1..] = y (B,N,T)
  float* y   = out + 1;

  // Workspace layout: repacked weights (6 MB) + per-block SSE partials
  float* Wp1 = (float*)d_ws;                       // 256*1024
  float* Wp2 = Wp1 + (size_t)NFEAT * HID;          // 1024*1024
  float* Wp3 = Wp2 + (size_t)HID * HID;            // 1024*256
  float* err_partials = Wp3 + (size_t)HID * NFEAT; // BATCH/ROWS floats

  // Repack weights into WMMA fragment order (deterministic, every call)
  {
    const int t1 = (HID  / 16) * (NFEAT / 4) * 32;   // 131072
    const int t2 = (HID  / 16) * (HID   / 4) * 32;   // 524288
    const int t3 = (NFEAT/ 16) * (HID   / 4) * 32;   // 131072
    repack_weights<<<(t1 + 255) / 256, 256, 0, stream>>>(W1, Wp1, NFEAT, HID);
    repack_weights<<<(t2 + 255) / 256, 256, 0, stream>>>(W2, Wp2, HID,   HID);
    repack_weights<<<(t3 + 255) / 256, 256, 0, stream>>>(W3, Wp3, HID,   NFEAT);
  }

  const int nblocks = BATCH / ROWS;         // 32 independent row tiles
  const size_t lds_bytes = (size_t)(3 * ROWS * SX + 2 * ROWS * SH) * sizeof(float); // ~178 KB

  heun_mlp_persistent<<<nblocks, THREADS, lds_bytes, stream>>>(
      Input, y_true, Wp1, b1, Wp2, b2, Wp3, b3, err_partials, y);

  finalize_err<<<1, 32, 0, stream>>>(
      err_partials, nblocks, out, 1.0f / (512.0f * 256.0f * 256.0f));
}